// Attr_Tokenizer_26877905338815
// MI455X (gfx1250) — compile-verified
//
#include <hip/hip_runtime.h>

typedef __attribute__((ext_vector_type(2))) float v2f;
typedef __attribute__((ext_vector_type(8))) float v8f;

#define WAVES_PER_WG 8
#define WG_SIZE (WAVES_PER_WG * 32)
#define GMAX_PAD 3728  // 61*61=3721 rounded up to multiple of 16

// score[m][n] = px*(-2gx) + py*(-2gy) + 1*|g|^2  ( = dist^2 - |p|^2 )
// A (16x4 f32): lane L<16 holds row L: v0=px, v1=py ; lane L>=16 holds row L-16: v0=1, v1=0
// B (4x16 f32): lane L<16 holds col L: v0=-2gx, v1=-2gy ; lane L>=16 holds col L-16: v0=|g|^2, v1=0
// D (16x16 f32): lane L<16, vgpr r -> M=r, N=L ; lane L>=16, vgpr r -> M=8+r, N=L-16
__global__ __launch_bounds__(WG_SIZE) void attr_tokenizer_wmma_kernel(
    const float* __restrict__ x, const float* __restrict__ y,
    const float* __restrict__ theta, const float* __restrict__ grid,
    int G, int Gpad, int N, int* __restrict__ out_idx, float* __restrict__ out_off)
{
    __shared__ float4 sg[GMAX_PAD];

    const int tid = threadIdx.x;

    // Cooperative preload: transformed grid in B-fragment-friendly layout.
    for (int i = tid; i < Gpad; i += WG_SIZE) {
        float gx = 0.0f, gy = 0.0f, g2 = 1e30f;  // padding never wins argmin
        if (i < G) {
            gx = grid[2 * i];
            gy = grid[2 * i + 1];
            g2 = gx * gx + gy * gy;
        }
        sg[i] = make_float4(-2.0f * gx, -2.0f * gy, g2, 0.0f);
    }
    __syncthreads();

    const int lane = tid & 31;
    const int wave = tid >> 5;
    const int qbase = (blockIdx.x * WAVES_PER_WG + wave) * 16;
    if (qbase >= N) return;

    // Lanes 0..15 own query row qbase+lane: center + rotate into anchor frame.
    const int qrow = qbase + (lane & 15);
    const float cx = x[2 * qrow]     - y[2 * qrow];
    const float cy = x[2 * qrow + 1] - y[2 * qrow + 1];
    const float phi = 1.57079632679489662f - theta[qrow];  // -(theta - pi/2)
    float sphi, cphi;
    sincosf(phi, &sphi, &cphi);
    const float rx = cx * cphi - cy * sphi;
    const float ry = cx * sphi + cy * cphi;

    const int half = lane >> 4;
    const int ncol = lane & 15;

    v2f afrag;
    afrag.x = half ? 1.0f : rx;
    afrag.y = half ? 0.0f : ry;

    float bestV[8];
    int   bestI[8];
#pragma unroll
    for (int r = 0; r < 8; ++r) { bestV[r] = 3.4e38f; bestI[r] = 0; }

    for (int t = 0; t < Gpad; t += 16) {
        // B fragment: one ds_load_b64 per lane.
        const float2 bv = ((const float2*)&sg[t + ncol])[half];
        v2f bfrag;
        bfrag.x = bv.x;
        bfrag.y = bv.y;

        v8f cz = {};
        v8f d = __builtin_amdgcn_wmma_f32_16x16x4_f32(
            /*neg_a=*/false, afrag, /*neg_b=*/false, bfrag,
            /*c_mod=*/(short)0, cz, /*reuse_a=*/false, /*reuse_b=*/false);

        const int n = t + ncol;
#pragma unroll
        for (int r = 0; r < 8; ++r) {
            const float v = d[r];
            const bool lt = v < bestV[r];   // strict: earlier column wins ties
            bestV[r] = lt ? v : bestV[r];
            bestI[r] = lt ? n : bestI[r];
        }
    }

    // Reduce (min, argidx) across the 16 lanes holding each row; ties -> smaller index.
#pragma unroll
    for (int r = 0; r < 8; ++r) {
        float v = bestV[r];
        int   i = bestI[r];
#pragma unroll
        for (int m = 1; m < 16; m <<= 1) {
            const float ov = __shfl_xor(v, m, 32);
            const int   oi = __shfl_xor(i, m, 32);
            if (ov < v || (ov == v && oi < i)) { v = ov; i = oi; }
        }
        bestV[r] = v;
        bestI[r] = i;
    }

    // Lane -> row mapping: lanes 0..7 write rows qbase+0..7, lanes 16..23 rows qbase+8..15.
    const int rsel = lane & 7;
    int bi = bestI[0];
#pragma unroll
    for (int r = 1; r < 8; ++r) {
        const bool p = (rsel == r);
        bi = p ? bestI[r] : bi;
    }

    // Fetch the writer row's rotated point (rows 8..15 live in lanes 8..15).
    const int srcl = rsel + 8 * half;
    const float wrx = __shfl(rx, srcl, 32);
    const float wry = __shfl(ry, srcl, 32);

    if ((lane & 15) < 8) {
        const int row = qbase + rsel + 8 * half;
        const float gx = grid[2 * bi];
        const float gy = grid[2 * bi + 1];
        out_idx[row] = bi;                 // int32 bits into first N slots
        out_off[2 * row]     = wrx - gx;
        out_off[2 * row + 1] = wry - gy;
    }
}

extern "C" void kernel_launch(void* const* d_in, const int* in_sizes, int n_in,
                              void* d_out, int out_size, void* d_ws, size_t ws_size,
                              hipStream_t stream) {
    const float* x     = (const float*)d_in[0];
    const float* y     = (const float*)d_in[1];
    const float* theta = (const float*)d_in[2];
    const float* grid  = (const float*)d_in[3];

    const int N = in_sizes[0] / 2;
    const int G = in_sizes[3] / 2;
    int Gpad = ((G + 15) / 16) * 16;
    if (Gpad > GMAX_PAD) Gpad = GMAX_PAD;

    int*   out_idx = (int*)d_out;           // first N elements: int32 indices
    float* out_off = (float*)d_out + N;     // next 2N: [N,2] offsets

    const int mtiles = (N + 15) / 16;
    const int blocks = (mtiles + WAVES_PER_WG - 1) / WAVES_PER_WG;
    attr_tokenizer_wmma_kernel<<<blocks, WG_SIZE, 0, stream>>>(
        x, y, theta, grid, G, Gpad, N, out_idx, out_off);
}